// PAMEncoder_22213570855360
// MI455X (gfx1250) — compile-verified
//
#include <hip/hip_runtime.h>
#include <hip/hip_bf16.h>
#include <math.h>

// Problem constants (match reference)
#define BB 128
#define LL 512
#define VV 21257
#define DD 300
#define AA 32

typedef __attribute__((ext_vector_type(2))) float v2f;
typedef __attribute__((ext_vector_type(8))) float v8f;

// ---------------- reductions (wave32) ----------------
__device__ __forceinline__ float blockSum256(float v, float* sm) {
#pragma unroll
    for (int o = 16; o > 0; o >>= 1) v += __shfl_xor(v, o, 32);
    int w = threadIdx.x >> 5;
    if ((threadIdx.x & 31) == 0) sm[w] = v;
    __syncthreads();
    float r = sm[0];
#pragma unroll
    for (int i = 1; i < 8; ++i) r += sm[i];
    __syncthreads();
    return r;
}

__device__ __forceinline__ float blockMax256(float v, float* sm) {
#pragma unroll
    for (int o = 16; o > 0; o >>= 1) v = fmaxf(v, __shfl_xor(v, o, 32));
    int w = threadIdx.x >> 5;
    if ((threadIdx.x & 31) == 0) sm[w] = v;
    __syncthreads();
    float r = sm[0];
#pragma unroll
    for (int i = 1; i < 8; ++i) r = fmaxf(r, sm[i]);
    __syncthreads();
    return r;
}

// ---------------- positional encoding ----------------
__global__ void pos_encoding_kernel(float* __restrict__ pos) {
    int idx = blockIdx.x * blockDim.x + threadIdx.x;
    if (idx >= LL * DD) return;
    int l = idx / DD, d = idx - l * DD;
    int i2 = d & ~1;  // 2*i
    float div = __expf((float)i2 * (-9.210340371976184f / (float)DD)); // -ln(1e4)/D
    float ang = (float)l * div;
    pos[idx] = (d & 1) ? cosf(ang) : sinf(ang);
}

// ---------------- embedding gather (float4 rows: 300/4 = 75) ----------------
__global__ void embed_kernel(const int* __restrict__ x,
                             const float4* __restrict__ emb4,
                             float4* __restrict__ e4) {
    const int ROW4 = DD / 4;
    int idx = blockIdx.x * blockDim.x + threadIdx.x;
    if (idx >= BB * LL * ROW4) return;
    int row = idx / ROW4, c = idx - row * ROW4;
    int xi = x[row];
    e4[idx] = emb4[(long long)xi * ROW4 + c];
}

// ---------------- WMMA f32 16x16x4 GEMM, 32x32 tile per wave ----------------
// C[b] = alpha * A[b] * op(B[b]) (+bias, +relu).
// NT: B is [N,K] row-major (i.e. B^T);  NN: B is [K,N] row-major.
// One wave owns a 32x32 C tile: 4 v8f accumulators, 4 WMMAs per K-step with
// software-pipelined (double-buffered) fragment loads -> 1 load per WMMA and
// independent accumulate chains so waits overlap the matrix pipe.
// Requirements: M % 32 == 0, K % 4 == 0 (guaranteed by caller); N edge handled
// with per-lane selects (EXEC stays all-ones around every WMMA).
template <bool NT, bool BIAS, bool RELU>
__global__ __launch_bounds__(256) void gemm_wmma_f32(
    const float* __restrict__ A, const float* __restrict__ Bm,
    const float* __restrict__ bias, float* __restrict__ C,
    int N, int K, int lda, int ldb, int ldc,
    long long sA, long long sB, long long sC,
    int mt2, int nt2, int nTiles, float alpha) {
    int wave = blockIdx.x * 8 + (int)(threadIdx.x >> 5);
    if (wave >= nTiles) return;            // wave-uniform branch
    int tpb = mt2 * nt2;
    int b = wave / tpb;
    int r = wave - b * tpb;
    int tm = r / nt2, tn = r - tm * nt2;
    int m0 = tm * 32, n0 = tn * 32;

    int lane = threadIdx.x & 31;
    int hh = lane >> 4;        // K-pair select for A/B frags, M-half select for C
    int lm = lane & 15;

    const float* Ab = A + (long long)b * sA;
    const float* Bb = Bm + (long long)b * sB;
    float* Cb = C + (long long)b * sC;

    int n_0 = n0 + lm;            // columns covered by b-frag 0
    int n_1 = n0 + 16 + lm;       // columns covered by b-frag 1
    bool ok0 = (n_0 < N), ok1 = (n_1 < N);
    int ns0 = ok0 ? n_0 : (N - 1);
    int ns1 = ok1 ? n_1 : (N - 1);

    const float* arow0 = Ab + (long long)(m0 + lm) * lda + 2 * hh;
    const float* arow1 = Ab + (long long)(m0 + 16 + lm) * lda + 2 * hh;
    const float* brow0 = Bb + (long long)ns0 * ldb + 2 * hh;  // NT only
    const float* brow1 = Bb + (long long)ns1 * ldb + 2 * hh;  // NT only

    auto loadB = [&](int k0, bool second) -> v2f {
        v2f bv;
        if (NT) {
            bv = *(const v2f*)((second ? brow1 : brow0) + k0);
        } else {
            int ka = k0 + 2 * hh;
            int ns = second ? ns1 : ns0;
            bv.x = Bb[(long long)ka * ldb + ns];
            bv.y = Bb[(long long)(ka + 1) * ldb + ns];
        }
        bool ok = second ? ok1 : ok0;
        bv.x = ok ? bv.x : 0.f;
        bv.y = ok ? bv.y : 0.f;
        return bv;
    };

    v8f acc00 = {0.f,0.f,0.f,0.f,0.f,0.f,0.f,0.f};
    v8f acc01 = acc00, acc10 = acc00, acc11 = acc00;

    // prologue: fragments for k=0
    v2f a0 = *(const v2f*)(arow0);
    v2f a1 = *(const v2f*)(arow1);
    v2f b0 = loadB(0, false);
    v2f b1 = loadB(0, true);

    int k0 = 0;
    for (; k0 + 4 < K; k0 += 4) {
        int kn = k0 + 4;
        // prefetch next K-step fragments before consuming current ones
        v2f a0n = *(const v2f*)(arow0 + kn);
        v2f a1n = *(const v2f*)(arow1 + kn);
        v2f b0n = loadB(kn, false);
        v2f b1n = loadB(kn, true);
        acc00 = __builtin_amdgcn_wmma_f32_16x16x4_f32(false, a0, false, b0, (short)0, acc00, false, false);
        acc01 = __builtin_amdgcn_wmma_f32_16x16x4_f32(false, a0, false, b1, (short)0, acc01, false, false);
        acc10 = __builtin_amdgcn_wmma_f32_16x16x4_f32(false, a1, false, b0, (short)0, acc10, false, false);
        acc11 = __builtin_amdgcn_wmma_f32_16x16x4_f32(false, a1, false, b1, (short)0, acc11, false, false);
        a0 = a0n; a1 = a1n; b0 = b0n; b1 = b1n;
    }
    // final K-step
    acc00 = __builtin_amdgcn_wmma_f32_16x16x4_f32(false, a0, false, b0, (short)0, acc00, false, false);
    acc01 = __builtin_amdgcn_wmma_f32_16x16x4_f32(false, a0, false, b1, (short)0, acc01, false, false);
    acc10 = __builtin_amdgcn_wmma_f32_16x16x4_f32(false, a1, false, b0, (short)0, acc10, false, false);
    acc11 = __builtin_amdgcn_wmma_f32_16x16x4_f32(false, a1, false, b1, (short)0, acc11, false, false);

    float bias0 = 0.f, bias1 = 0.f;
    if (BIAS) {
        bias0 = ok0 ? bias[ns0] : 0.f;
        bias1 = ok1 ? bias[ns1] : 0.f;
    }
#pragma unroll
    for (int rr = 0; rr < 8; ++rr) {
        int mA = m0 + rr + 8 * hh;   // C layout: VGPR rr -> M = rr + 8*(lane/16)
        int mB = mA + 16;
        float v00 = acc00[rr] * alpha, v01 = acc01[rr] * alpha;
        float v10 = acc10[rr] * alpha, v11 = acc11[rr] * alpha;
        if (BIAS) { v00 += bias0; v01 += bias1; v10 += bias0; v11 += bias1; }
        if (RELU) {
            v00 = fmaxf(v00, 0.f); v01 = fmaxf(v01, 0.f);
            v10 = fmaxf(v10, 0.f); v11 = fmaxf(v11, 0.f);
        }
        if (ok0) Cb[(long long)mA * ldc + n_0] = v00;
        if (ok1) Cb[(long long)mA * ldc + n_1] = v01;
        if (ok0) Cb[(long long)mB * ldc + n_0] = v10;
        if (ok1) Cb[(long long)mB * ldc + n_1] = v11;
    }
}

// ---------------- masked softmax over rows (query axis pre-transposed) ----------------
// dst[b, j, q] = softmax_q( mask(b,j,q) ? src_row[q] : -1e7 )
// src row = src + b*srcStrideB + j*L  (srcStrideB=0 for batch-shared symmetric S0)
__global__ __launch_bounds__(256) void masked_softmax_kernel(
    const float* __restrict__ src, long long srcStrideB,
    float* __restrict__ dst, const unsigned char* __restrict__ mask) {
    int j = blockIdx.x, b = blockIdx.y;
    const float* srow = src + (long long)b * srcStrideB + (long long)j * LL;
    float* drow = dst + ((long long)b * LL + j) * LL;
    const unsigned char* mrow = mask + (long long)b * LL;
    bool mj = mrow[j] != 0;
    int t = threadIdx.x;
    int q0 = t, q1 = t + 256;
    float v0 = (mj && mrow[q0]) ? srow[q0] : -1e7f;
    float v1 = (mj && mrow[q1]) ? srow[q1] : -1e7f;
    __shared__ float sm[8];
    float mx = blockMax256(fmaxf(v0, v1), sm);
    float e0 = __expf(v0 - mx), e1 = __expf(v1 - mx);
    float s = blockSum256(e0 + e1, sm);
    float inv = 1.f / s;
    drow[q0] = e0 * inv;
    drow[q1] = e1 * inv;
}

// ---------------- layernorm (unbiased var, /(std+eps)) + residual, in-place ----------
__global__ __launch_bounds__(256) void ln_residual_kernel(
    float* __restrict__ h, const float* __restrict__ e,
    const float* __restrict__ a2, const float* __restrict__ b2) {
    long long row = blockIdx.x;
    float* hr = h + row * DD;
    const float* er = e + row * DD;
    int t = threadIdx.x;
    float x0 = (t < DD) ? hr[t] : 0.f;
    float x1 = (t + 256 < DD) ? hr[t + 256] : 0.f;
    __shared__ float sm[8];
    float s = blockSum256(x0 + x1, sm);
    float mean = s / (float)DD;
    float d0 = (t < DD) ? (x0 - mean) : 0.f;
    float d1 = (t + 256 < DD) ? (x1 - mean) : 0.f;
    float ss = blockSum256(d0 * d0 + d1 * d1, sm);
    float var = ss / (float)(DD - 1);
    float inv = 1.f / (sqrtf(var) + 1e-6f);
    if (t < DD) hr[t] = er[t] + a2[t] * d0 * inv + b2[t];
    if (t + 256 < DD) hr[t + 256] = er[t + 256] + a2[t + 256] * d1 * inv + b2[t + 256];
}

// ---------------- masked pool over L + L2 normalize ----------------
__global__ __launch_bounds__(256) void pool_normalize_kernel(
    const float* __restrict__ h, const unsigned char* __restrict__ mask,
    float* __restrict__ out) {
    int b = blockIdx.x, t = threadIdx.x;
    const unsigned char* mrow = mask + (long long)b * LL;
    const float* hb = h + (long long)b * LL * DD;
    float acc0 = 0.f, acc1 = 0.f;
    for (int l = 0; l < LL; ++l) {
        float g = mrow[l] ? 1.f : 0.f;
        if (t < DD) acc0 += g * hb[(long long)l * DD + t];
        if (t + 256 < DD) acc1 += g * hb[(long long)l * DD + t + 256];
    }
    float sq = 0.f;
    if (t < DD) sq += acc0 * acc0;
    if (t + 256 < DD) sq += acc1 * acc1;
    __shared__ float sm[8];
    float tot = blockSum256(sq, sm);
    float inv = 1.f / sqrtf(tot);
    if (t < DD) out[(long long)b * DD + t] = acc0 * inv;
    if (t + 256 < DD) out[(long long)b * DD + t + 256] = acc1 * inv;
}

// ---------------- launcher ----------------
extern "C" void kernel_launch(void* const* d_in, const int* in_sizes, int n_in,
                              void* d_out, int out_size, void* d_ws, size_t ws_size,
                              hipStream_t stream) {
    (void)in_sizes; (void)n_in; (void)out_size; (void)ws_size;
    const int*           x      = (const int*)d_in[0];
    const unsigned char* mask   = (const unsigned char*)d_in[1];   // jnp bool -> 1 byte
    const float*         emb    = (const float*)d_in[2];
    const float*         pam_w  = (const float*)d_in[3];
    const float*         pam_b  = (const float*)d_in[4];
    const float*         sam_wq = (const float*)d_in[5];
    const float*         sam_bq = (const float*)d_in[6];
    const float*         sam_wk = (const float*)d_in[7];
    const float*         sam_bk = (const float*)d_in[8];
    const float*         proj_w = (const float*)d_in[9];
    const float*         proj_b = (const float*)d_in[10];
    const float*         ln_a   = (const float*)d_in[11];
    const float*         ln_b   = (const float*)d_in[12];
    float* out = (float*)d_out;

    // workspace carve-up (floats)
    float* ws = (float*)d_ws;
    long long off = 0;
    float* pos  = ws + off; off += (long long)LL * DD;        // [L,D]
    float* S0   = ws + off; off += (long long)LL * LL;        // [L,L] shared, symmetric
    float* e    = ws + off; off += (long long)BB * LL * DD;   // embeddings
    float* bufA = ws + off; off += (long long)BB * LL * DD;   // attn outs
    float* bufB = ws + off; off += (long long)BB * LL * DD;   // pam_out / c / h
    float* qb   = ws + off; off += (long long)BB * LL * AA;
    float* kb   = ws + off; off += (long long)BB * LL * AA;
    float* S    = ws + off; off += (long long)BB * LL * LL;   // SAM scores^T
    float* P    = ws + off; off += (long long)BB * LL * LL;   // probs^T

    const float inv_s1 = 1.0f / powf((float)DD, 0.25f);
    const float inv_s2 = 1.0f / powf((float)AA, 0.25f);

    // 1) positional encoding
    pos_encoding_kernel<<<(LL * DD + 255) / 256, 256, 0, stream>>>(pos);
    // 2) embedding gather
    embed_kernel<<<(BB * LL * (DD / 4) + 255) / 256, 256, 0, stream>>>(
        x, (const float4*)emb, (float4*)e);

    // 3) S0 = pos * pos^T * inv_s1  (M=N=512, K=300, NT, batch-shared)
    {
        int mt2 = LL / 32, nt2 = LL / 32, tiles = mt2 * nt2;
        gemm_wmma_f32<true, false, false><<<(tiles + 7) / 8, 256, 0, stream>>>(
            pos, pos, nullptr, S0, LL, DD, DD, DD, LL, 0, 0, 0, mt2, nt2, tiles, inv_s1);
    }
    // 4) PAM softmax over query axis: P[b,k,q] (S0 symmetric -> row read)
    masked_softmax_kernel<<<dim3(LL, BB), 256, 0, stream>>>(S0, 0, P, mask);
    // 5) bufA[b] = P[b] * e[b]   (M=512, N=300, K=512, NN, batched)
    {
        int mt2 = LL / 32, nt2 = (DD + 31) / 32, tiles = mt2 * nt2 * BB;
        gemm_wmma_f32<false, false, false><<<(tiles + 7) / 8, 256, 0, stream>>>(
            P, e, nullptr, bufA, DD, LL, LL, DD, DD,
            (long long)LL * LL, (long long)LL * DD, (long long)LL * DD,
            mt2, nt2, tiles, 1.0f);
    }
    // 6) bufB = relu(bufA * pam_w^T + pam_b)  (M=B*L, N=300, K=300, NT)
    {
        int mt2 = (BB * LL) / 32, nt2 = (DD + 31) / 32, tiles = mt2 * nt2;
        gemm_wmma_f32<true, true, true><<<(tiles + 7) / 8, 256, 0, stream>>>(
            bufA, pam_w, pam_b, bufB, DD, DD, DD, DD, DD, 0, 0, 0, mt2, nt2, tiles, 1.0f);
    }
    // 7/8) q,k = relu(bufB * w^T + b)  (N=32, K=300, NT)
    {
        int mt2 = (BB * LL) / 32, nt2 = AA / 32, tiles = mt2 * nt2;
        gemm_wmma_f32<true, true, true><<<(tiles + 7) / 8, 256, 0, stream>>>(
            bufB, sam_wq, sam_bq, qb, AA, DD, DD, DD, AA, 0, 0, 0, mt2, nt2, tiles, 1.0f);
        gemm_wmma_f32<true, true, true><<<(tiles + 7) / 8, 256, 0, stream>>>(
            bufB, sam_wk, sam_bk, kb, AA, DD, DD, DD, AA, 0, 0, 0, mt2, nt2, tiles, 1.0f);
    }
    // 9) S[b] = kb[b] * qb[b]^T * inv_s2  -> S[b,j,i] = scores[b,i,j]  (NT, K=32)
    {
        int mt2 = LL / 32, nt2 = LL / 32, tiles = mt2 * nt2 * BB;
        gemm_wmma_f32<true, false, false><<<(tiles + 7) / 8, 256, 0, stream>>>(
            kb, qb, nullptr, S, LL, AA, AA, AA, LL,
            (long long)LL * AA, (long long)LL * AA, (long long)LL * LL,
            mt2, nt2, tiles, inv_s2);
    }
    // 10) SAM softmax over query axis (rows of S^T)
    masked_softmax_kernel<<<dim3(LL, BB), 256, 0, stream>>>(
        S, (long long)LL * LL, P, mask);
    // 11) bufA[b] = P[b] * bufB[b]  (sam_out; NN, K=512)
    {
        int mt2 = LL / 32, nt2 = (DD + 31) / 32, tiles = mt2 * nt2 * BB;
        gemm_wmma_f32<false, false, false><<<(tiles + 7) / 8, 256, 0, stream>>>(
            P, bufB, nullptr, bufA, DD, LL, LL, DD, DD,
            (long long)LL * LL, (long long)LL * DD, (long long)LL * DD,
            mt2, nt2, tiles, 1.0f);
    }
    // 12) bufB = relu(bufA * proj_w^T + proj_b)
    {
        int mt2 = (BB * LL) / 32, nt2 = (DD + 31) / 32, tiles = mt2 * nt2;
        gemm_wmma_f32<true, true, true><<<(tiles + 7) / 8, 256, 0, stream>>>(
            bufA, proj_w, proj_b, bufB, DD, DD, DD, DD, DD, 0, 0, 0, mt2, nt2, tiles, 1.0f);
    }
    // 13) bufB = e + layernorm(bufB)   (in-place per row)
    ln_residual_kernel<<<BB * LL, 256, 0, stream>>>(bufB, e, ln_a, ln_b);
    // 14) masked pool over L + L2 normalize -> d_out [B,D]
    pool_normalize_kernel<<<BB, 256, 0, stream>>>(bufB, mask, out);
}